// WadaIN_71768903516410
// MI455X (gfx1250) — compile-verified
//
#include <hip/hip_runtime.h>
#include <hip/hip_bf16.h>

typedef __attribute__((ext_vector_type(2))) float v2f;
typedef __attribute__((ext_vector_type(8))) float v8f;

#define B_    16
#define CIN   256
#define COUT  256
#define KW    5
#define SPK_  128
#define T_    8192
#define TB    512          // time columns per block
#define XS_STRIDE 520      // 516 used; pad so row+2 hits disjoint bank groups
#define AS_STRIDE 82       // 80 used; even -> b64 A-frag loads, 18L%64 distinct banks
#define NXV   16           // float2 interior x slots per thread (16*256 = 4096 = 16x512/2)
#define NA    5            // A elements per thread (16*80/256)

// ---------------- kernel 1: style = c_trg @ style_w^T + style_b ----------------
__global__ void style_kernel(const float* __restrict__ c_trg,
                             const float* __restrict__ style_w,
                             const float* __restrict__ style_b,
                             float* __restrict__ s) {
  int b = blockIdx.x;
  int i = threadIdx.x;
  const float* c = c_trg + b * SPK_;
  const float* w = style_w + i * SPK_;
  float acc = style_b[i];
  #pragma unroll 8
  for (int j = 0; j < SPK_; ++j) acc += c[j] * w[j];
  s[b * CIN + i] = acc;
}

// ---------------- kernel 2: modulate + demodulate weights ----------------
__global__ void wmod_kernel(const float* __restrict__ weight,
                            const float* __restrict__ s,
                            float* __restrict__ wmod) {
  int o = blockIdx.x;
  int b = blockIdx.y;
  int i = threadIdx.x;
  float si = s[b * CIN + i];
  const float* wp = weight + (size_t)(o * CIN + i) * KW;
  float w[KW];
  float ss = 0.f;
  #pragma unroll
  for (int k = 0; k < KW; ++k) { w[k] = wp[k] * si; ss += w[k] * w[k]; }

  __shared__ float red[CIN];
  red[i] = ss;
  __syncthreads();
  for (int st = CIN / 2; st > 0; st >>= 1) {
    if (i < st) red[i] += red[i + st];
    __syncthreads();
  }
  float demod = rsqrtf(red[0] + 1e-8f);
  float* op = wmod + (size_t)(b * COUT + o) * (CIN * KW) + i * KW;
  #pragma unroll
  for (int k = 0; k < KW; ++k) op[k] = w[k] * demod;
}

// ---------------- kernel 3: implicit-im2col conv GEMM via fp32 WMMA ----------------
// Register-software-pipelined: prefetch chunk ic+1 into VGPRs while chunk ic's
// 80 WMMAs execute; regs -> LDS between barriers (single LDS buffer).
__global__ __launch_bounds__(256) void conv_wmma_kernel(
    const float* __restrict__ x,
    const float* __restrict__ wmod,
    float* __restrict__ out) {
  const int tb  = blockIdx.x;        // time block (512 cols)
  const int ot  = blockIdx.y;        // out-channel tile (16 rows)
  const int b   = blockIdx.z;        // batch
  const int tid = threadIdx.x;
  const int lane = tid & 31;         // wave32
  const int wv   = tid >> 5;         // 8 waves
  const int hl   = lane >> 4;        // half-wave select (K 0/1 vs 2/3)
  const int ln   = lane & 15;        // M (A) / N (B,C) index

  __shared__ float xs[16 * XS_STRIDE];   // 16 channels x (512+4) cols
  __shared__ float as[16 * AS_STRIDE];   // 16 outs x 80 (q = k*16 + i)

  const int tb0 = tb * TB;
  const float* xb = x + (size_t)b * CIN * T_;
  const float* wb = wmod + (size_t)(b * COUT + ot * 16) * (CIN * KW);

  // A staging map (no divisions): thread -> (o = tid>>4, i = tid&15, k = j)
  const int ao = tid >> 4;
  const int ai = tid & 15;
  // x edge map: threads 0..63 -> (row = tid>>2, e = tid&3), cols {0,1,514,515}
  const int er   = tid >> 2;
  const int ee   = tid & 3;
  const int ecol = (ee < 2) ? ee : (512 + ee);
  int et = tb0 + ecol - 2;
  et = (et < 0) ? -et : et;
  et = (et >= T_) ? (2 * T_ - 2 - et) : et;

  v2f  xr2[NXV];   // interior x prefetch (float2, coalesced b64)
  float xe = 0.f;  // edge x prefetch (threads 0..63)
  float ar[NA];    // A prefetch

  // ---- prefetch chunk 0 ----
  {
    const float* xc = xb;                                // i0 = 0
    #pragma unroll
    for (int j = 0; j < NXV; ++j) {
      int idx = tid + j * 256;
      int r = idx >> 8;            // 256 float2 per row
      int c2 = idx & 255;
      xr2[j] = *(const v2f*)(xc + (size_t)r * T_ + tb0 + 2 * c2);
    }
    if (tid < 64) xe = xc[(size_t)er * T_ + et];
    #pragma unroll
    for (int j = 0; j < NA; ++j)
      ar[j] = wb[(size_t)ao * (CIN * KW) + ai * KW + j];
  }

  v8f acc[4] = {};

  for (int ic = 0; ic < CIN / 16; ++ic) {
    __syncthreads();                 // previous chunk's compute done
    // ---- regs -> LDS ----
    #pragma unroll
    for (int j = 0; j < NXV; ++j) {
      int idx = tid + j * 256;
      int r = idx >> 8;
      int c2 = idx & 255;
      *(v2f*)(&xs[r * XS_STRIDE + 2 + 2 * c2]) = xr2[j];
    }
    if (tid < 64) xs[er * XS_STRIDE + ecol] = xe;
    #pragma unroll
    for (int j = 0; j < NA; ++j)
      as[ao * AS_STRIDE + j * 16 + ai] = ar[j];   // q = k*16 + i
    __syncthreads();

    // ---- issue prefetch for next chunk (hidden behind WMMAs) ----
    if (ic + 1 < CIN / 16) {
      const int i0n = (ic + 1) * 16;
      const float* xc = xb + (size_t)i0n * T_;
      #pragma unroll
      for (int j = 0; j < NXV; ++j) {
        int idx = tid + j * 256;
        int r = idx >> 8;
        int c2 = idx & 255;
        xr2[j] = *(const v2f*)(xc + (size_t)r * T_ + tb0 + 2 * c2);
      }
      if (tid < 64) xe = xc[(size_t)er * T_ + et];
      const float* wc = wb + i0n * KW;
      #pragma unroll
      for (int j = 0; j < NA; ++j)
        ar[j] = wc[(size_t)ao * (CIN * KW) + ai * KW + j];
    }

    // ---- 20 K-steps x 4 N-tiles of fp32 WMMA ----
    const int wt0 = wv * 64;
    #pragma unroll 4
    for (int q0 = 0; q0 < 80; q0 += 4) {
      const int k  = q0 >> 4;        // kernel tap (constant within K-step)
      const int ib = q0 & 15;        // base channel within chunk
      v2f a;                         // A frag: lane ln = row M; K = hl*2, hl*2+1
      a.x = as[ln * AS_STRIDE + q0 + hl * 2 + 0];
      a.y = as[ln * AS_STRIDE + q0 + hl * 2 + 1];
      const int r0 = ib + hl * 2;
      const float* xr = &xs[r0 * XS_STRIDE];
      #pragma unroll
      for (int nt = 0; nt < 4; ++nt) {
        const int col = wt0 + nt * 16 + ln + k;
        v2f bv;                      // B frag: lane ln = col N; K = hl*2, hl*2+1
        bv.x = xr[col];
        bv.y = xr[XS_STRIDE + col];
        acc[nt] = __builtin_amdgcn_wmma_f32_16x16x4_f32(
            false, a, false, bv, (short)0, acc[nt], false, false);
      }
    }
  }

  // ---- epilogue: leaky-ReLU + store (VGPR r -> M = r + hl*8, lane -> N) ----
  float* ob = out + ((size_t)b * COUT + ot * 16) * T_ + tb0 + wv * 64;
  #pragma unroll
  for (int nt = 0; nt < 4; ++nt) {
    #pragma unroll
    for (int r = 0; r < 8; ++r) {
      const int M = r + hl * 8;
      float v = acc[nt][r];
      v = (v > 0.f) ? v : 0.2f * v;
      ob[(size_t)M * T_ + nt * 16 + ln] = v;
    }
  }
}

// ---------------- kernel 4: append c_trg to output tuple ----------------
__global__ void tail_kernel(const float* __restrict__ c_trg, float* __restrict__ out) {
  int i = blockIdx.x * 256 + threadIdx.x;
  if (i < B_ * SPK_) out[(size_t)B_ * COUT * T_ + i] = c_trg[i];
}

extern "C" void kernel_launch(void* const* d_in, const int* in_sizes, int n_in,
                              void* d_out, int out_size, void* d_ws, size_t ws_size,
                              hipStream_t stream) {
  const float* x       = (const float*)d_in[0];   // [16,256,8192]
  const float* c_trg   = (const float*)d_in[1];   // [16,128]
  const float* style_w = (const float*)d_in[2];   // [256,128]
  const float* style_b = (const float*)d_in[3];   // [256]
  const float* weight  = (const float*)d_in[4];   // [1,256,256,5]
  float* out = (float*)d_out;

  float* s    = (float*)d_ws;            // 16*256 floats
  float* wmod = s + B_ * CIN;            // 16*256*1280 floats (~21 MB)

  style_kernel<<<dim3(B_), dim3(CIN), 0, stream>>>(c_trg, style_w, style_b, s);
  wmod_kernel<<<dim3(COUT, B_), dim3(CIN), 0, stream>>>(weight, s, wmod);
  conv_wmma_kernel<<<dim3(T_ / TB, COUT / 16, B_), dim3(256), 0, stream>>>(x, wmod, out);
  tail_kernel<<<dim3((B_ * SPK_ + 255) / 256), dim3(256), 0, stream>>>(c_trg, out);
}